// TaylorRule3Var_55319178772923
// MI455X (gfx1250) — compile-verified
//
#include <hip/hip_runtime.h>
#include <hip/hip_bf16.h>

typedef __attribute__((ext_vector_type(2))) float v2f;
typedef __attribute__((ext_vector_type(8))) float v8f;

#define N_PRE  8192   // j (columns, x)
#define N_POST 8192   // i (rows, y)

// dW[i,j] = sum_c w^c * S_c,  S_c = sum_b y^b * Q[b,c](x)
// S_c computed per 16x16 tile with V_WMMA_F32_16X16X4_F32 (K=4, b-dim padded).
__global__ __launch_bounds__(256) void taylor3_wmma_kernel(
    const float* __restrict__ x, const float* __restrict__ y,
    const float* __restrict__ w, const float* __restrict__ coeffs,
    float* __restrict__ out)
{
    const int lane = threadIdx.x & 31;
    const int wave = threadIdx.x >> 5;
    const int n    = lane & 15;        // column within tile / M row for A
    const bool hi  = lane >= 16;

    const int jt = blockIdx.x * 8 + wave;   // 512 j-tiles total
    const int it = blockIdx.y;              // 512 i-tiles
    const int j0 = jt << 4;
    const int i0 = it << 4;

    // ---- A matrix (16x4 f32): y-powers. Layout: lane%16 = M row;
    // VGPR0 holds K = (hi?2:0), VGPR1 holds K = (hi?3:1); K=3 is zero pad.
    const float yv = y[i0 + n];
    v2f A;
    A.x = hi ? yv * yv : 1.0f;   // K=2 : K=0
    A.y = hi ? 0.0f    : yv;     // K=3 (pad) : K=1

    // ---- Per-lane column data for B: Q[b][c] = C0 + x*C1 + x2*C2.
    // coeffs flat index = a*9 + b*3 + c (uniform offsets -> scalar loads).
    const float xv = x[j0 + n];
    const float x2 = xv * xv;
    float q[3][3];
#pragma unroll
    for (int b = 0; b < 3; ++b)
#pragma unroll
        for (int c = 0; c < 3; ++c)
            q[b][c] = coeffs[b*3 + c]
                    + xv * coeffs[9  + b*3 + c]
                    + x2 * coeffs[18 + b*3 + c];

    // ---- S_c = A(16x4) x B_c(4x16) via WMMA. B layout mirrors A:
    // VGPR0: row K=(hi?2:0), VGPR1: row K=(hi?3:1); N = lane%16.
    const v8f Z = {0.f, 0.f, 0.f, 0.f, 0.f, 0.f, 0.f, 0.f};
    v8f S[3];
#pragma unroll
    for (int c = 0; c < 3; ++c) {
        v2f B;
        B.x = hi ? q[2][c] : q[0][c];
        B.y = hi ? 0.0f    : q[1][c];
        S[c] = __builtin_amdgcn_wmma_f32_16x16x4_f32(
            /*neg_a=*/false, A, /*neg_b=*/false, B,
            /*c_mod=*/(short)0, Z, /*reuse_a=*/false, /*reuse_b=*/false);
    }

    // ---- Elementwise combine with w powers. C/D layout: VGPR r maps to
    // row i0 + r + (hi?8:0), column j0 + lane%16. Streamed (non-temporal):
    // w and dW are 256 MB each, touched exactly once.
    const int    rbase = i0 + (hi ? 8 : 0);
    const size_t col   = (size_t)(j0 + n);
#pragma unroll
    for (int r = 0; r < 8; ++r) {
        size_t idx = (size_t)(rbase + r) * (size_t)N_PRE + col;
        float wv = __builtin_nontemporal_load(&w[idx]);
        float d  = S[0][r] + wv * S[1][r] + (wv * wv) * S[2][r];
        __builtin_nontemporal_store(d, &out[idx]);
    }
}

extern "C" void kernel_launch(void* const* d_in, const int* in_sizes, int n_in,
                              void* d_out, int out_size, void* d_ws, size_t ws_size,
                              hipStream_t stream) {
    (void)in_sizes; (void)n_in; (void)out_size; (void)d_ws; (void)ws_size;
    const float* x      = (const float*)d_in[0];   // [8192]
    const float* y      = (const float*)d_in[1];   // [8192]
    const float* w      = (const float*)d_in[2];   // [8192*8192]
    const float* coeffs = (const float*)d_in[3];   // [27]
    float* out = (float*)d_out;

    dim3 block(256, 1, 1);                       // 8 waves, 1 tile each
    dim3 grid(N_PRE / 16 / 8, N_POST / 16, 1);   // (64, 512)
    taylor3_wmma_kernel<<<grid, block, 0, stream>>>(x, y, w, coeffs, out);
}